// LightGCN_48464410968713
// MI455X (gfx1250) — compile-verified
//
#include <hip/hip_runtime.h>
#include <hip/hip_bf16.h>
#include <stdint.h>

#define EMBED_DIM 64

#if defined(__HIP_DEVICE_COMPILE__) && __has_builtin(__builtin_amdgcn_global_load_async_to_lds_b64)
#define HAS_ASYNC_LDS 1
#else
#define HAS_ASYNC_LDS 0
#endif

#if HAS_ASYNC_LDS
typedef int v2i_t __attribute__((vector_size(8)));
typedef __attribute__((address_space(1))) v2i_t* g_v2i_ptr;   // global (AS1) -> v2i
typedef __attribute__((address_space(3))) v2i_t* l_v2i_ptr;   // LDS (AS3)   -> v2i
#endif

// ---------------------------------------------------------------- utilities
__global__ void zero_int_kernel(int* __restrict__ p, int n) {
    int i = blockIdx.x * blockDim.x + threadIdx.x;
    if (i < n) p[i] = 0;
}

// ------------------------------------------------------- degree counting
__global__ void count_deg_kernel(const int* __restrict__ user_ids,
                                 const int* __restrict__ item_ids,
                                 int* __restrict__ deg, int n_users, int E) {
    int e = blockIdx.x * blockDim.x + threadIdx.x;
    if (e >= E) return;
    if (e + 4096 < E) {
        __builtin_prefetch(user_ids + e + 4096, 0, 0);
        __builtin_prefetch(item_ids + e + 4096, 0, 0);
    }
    int u = user_ids[e];
    int v = item_ids[e] + n_users;
    atomicAdd(&deg[u], 1);
    atomicAdd(&deg[v], 1);
}

// ------------------------------------------------------- exclusive scan
__global__ __launch_bounds__(1024) void scan_block_kernel(const int* __restrict__ deg,
                                                          int* __restrict__ offs,
                                                          int* __restrict__ bsums, int n) {
    __shared__ int sh[1024];
    const int tid = threadIdx.x;
    const int gid = blockIdx.x * 1024 + tid;
    int v = (gid < n) ? deg[gid] : 0;
    sh[tid] = v;
    __syncthreads();
    for (int off = 1; off < 1024; off <<= 1) {
        int t = (tid >= off) ? sh[tid - off] : 0;
        __syncthreads();
        sh[tid] += t;
        __syncthreads();
    }
    int incl = sh[tid];
    if (gid < n) offs[gid] = incl - v;          // exclusive within block
    if (tid == 1023) bsums[blockIdx.x] = incl;  // block total
}

__global__ void scan_sums_kernel(int* __restrict__ bsums, int nb) {
    if (blockIdx.x == 0 && threadIdx.x == 0) {
        int run = 0;
        for (int i = 0; i < nb; ++i) { int t = bsums[i]; bsums[i] = run; run += t; }
    }
}

__global__ void finalize_kernel(int* __restrict__ offs, const int* __restrict__ bsums,
                                const int* __restrict__ deg, float* __restrict__ dinv,
                                int n, int total) {
    int i = blockIdx.x * blockDim.x + threadIdx.x;
    if (i < n) {
        offs[i] += bsums[i >> 10];
        int dg = deg[i];
        dinv[i] = (dg > 0) ? rsqrtf((float)dg) : 0.0f;
    }
    if (i == 0) offs[n] = total;
}

// ------------------------------------------------------- CSR bucket fill
__global__ void fill_csr_kernel(const int* __restrict__ user_ids,
                                const int* __restrict__ item_ids,
                                const int* __restrict__ offs, int* __restrict__ fill,
                                int* __restrict__ nbr, int n_users, int E) {
    int e = blockIdx.x * blockDim.x + threadIdx.x;
    if (e >= E) return;
    if (e + 4096 < E) {
        __builtin_prefetch(user_ids + e + 4096, 0, 0);
        __builtin_prefetch(item_ids + e + 4096, 0, 0);
    }
    int u = user_ids[e];
    int v = item_ids[e] + n_users;
    int pu = atomicAdd(&fill[u], 1);
    nbr[offs[u] + pu] = v;
    int pv = atomicAdd(&fill[v], 1);
    nbr[offs[v] + pv] = u;
}

// ------------------------------------------------------- acc = emb0
__global__ void init_out_kernel(const float* __restrict__ user_emb,
                                const float* __restrict__ item_emb,
                                float* __restrict__ out, int n_users, int n_nodes) {
    int i = blockIdx.x * blockDim.x + threadIdx.x;
    int total = n_nodes * EMBED_DIM;
    if (i >= total) return;
    int node = i >> 6;
    out[i] = (node < n_users) ? user_emb[i]
                              : item_emb[(size_t)i - (size_t)n_users * EMBED_DIM];
}

// ------------------------------------------------------- gather SpMM layer
// One wave (32 lanes) per destination node; 32 lanes x float2 = 64 dims.
// Neighbor rows (256B each) are gathered into a 4-deep LDS ring via CDNA5
// async global->LDS loads staged 3 ahead; in-order async completion lets
// s_wait_asynccnt with a constant immediate retire exactly the oldest stage.
__global__ __launch_bounds__(256) void layer_kernel(
    const float* __restrict__ in_u, const float* __restrict__ in_i,
    float* __restrict__ out, float* __restrict__ acc,
    const int* __restrict__ offs, const int* __restrict__ nbr,
    const float* __restrict__ dinv, int n_users, int n_nodes) {
    __shared__ alignas(16) float lds[8][4][EMBED_DIM];   // 8 waves x 4-deep ring
    const int lane = threadIdx.x & 31;
    const int wslot = threadIdx.x >> 5;
    const int node = blockIdx.x * 8 + wslot;
    if (node >= n_nodes) return;

    const int beg = offs[node];
    const int end = offs[node + 1];
    const int deg = end - beg;
    const float wd = dinv[node];

    float sx = 0.0f, sy = 0.0f;

#if HAS_ASYNC_LDS
    auto stage = [&](int idx) {   // idx in [0, deg): gather row nbr[beg+idx] into buffer idx&3
        int nb = nbr[beg + idx];
        const float* row = (nb < n_users)
                               ? (in_u + (size_t)nb * EMBED_DIM)
                               : (in_i + (size_t)(nb - n_users) * EMBED_DIM);
        __builtin_amdgcn_global_load_async_to_lds_b64(
            (g_v2i_ptr)(row + lane * 2),
            (l_v2i_ptr)&lds[wslot][idx & 3][lane * 2],
            0, 0);
    };
    auto consume = [&](int idx) {
        int nb = nbr[beg + idx];
        float wn = dinv[nb];
        float ex = lds[wslot][idx & 3][lane * 2];
        float ey = lds[wslot][idx & 3][lane * 2 + 1];
        sx = fmaf(wn, ex, sx);
        sy = fmaf(wn, ey, sy);
    };

    // prologue: stage up to 3 ahead
    const int pre = (deg < 3) ? deg : 3;
    for (int j = 0; j < pre; ++j) stage(j);

    int idx = 0;
    // steady state: 4 in flight; wait <=3 retires the oldest (in-order)
    for (; idx + 3 < deg; ++idx) {
        stage(idx + 3);
        asm volatile("s_wait_asynccnt 0x3" ::: "memory");
        consume(idx);
    }
    // epilogue: drain
    asm volatile("s_wait_asynccnt 0x0" ::: "memory");
    for (; idx < deg; ++idx) consume(idx);
#else
    for (int k = beg; k < end; ++k) {
        int nb = nbr[k];
        const float* row = (nb < n_users)
                               ? (in_u + (size_t)nb * EMBED_DIM)
                               : (in_i + (size_t)(nb - n_users) * EMBED_DIM);
        float wn = dinv[nb];
        sx = fmaf(wn, row[lane * 2], sx);
        sy = fmaf(wn, row[lane * 2 + 1], sy);
    }
#endif

    size_t o = (size_t)node * EMBED_DIM + (size_t)(lane * 2);
    float ox = wd * sx, oy = wd * sy;
    out[o] = ox;
    out[o + 1] = oy;
    acc[o] += ox;   // only this wave touches row `node` -> no atomics needed
    acc[o + 1] += oy;
}

__global__ void scale_kernel(float* __restrict__ p, float s, int n) {
    int i = blockIdx.x * blockDim.x + threadIdx.x;
    if (i < n) p[i] *= s;
}

// ---------------------------------------------------------------- launch
extern "C" void kernel_launch(void* const* d_in, const int* in_sizes, int n_in,
                              void* d_out, int out_size, void* d_ws, size_t ws_size,
                              hipStream_t stream) {
    const float* user_emb = (const float*)d_in[0];
    const float* item_emb = (const float*)d_in[1];
    const int* user_ids = (const int*)d_in[2];
    const int* item_ids = (const int*)d_in[3];

    const int n_users = in_sizes[0] / EMBED_DIM;
    const int n_items = in_sizes[1] / EMBED_DIM;
    const int E = in_sizes[2];
    const int n_nodes = n_users + n_items;
    const int twoE = 2 * E;
    const int n_elems = n_nodes * EMBED_DIM;

    // workspace carve-out
    char* ws = (char*)d_ws;
    size_t off = 0;
    auto alloc = [&](size_t bytes) -> void* {
        void* p = ws + off;
        off += (bytes + 255) & ~(size_t)255;
        return p;
    };
    int* deg = (int*)alloc((size_t)n_nodes * 4);
    int* fill = (int*)alloc((size_t)n_nodes * 4);
    int* offs = (int*)alloc((size_t)(n_nodes + 1) * 4);
    int* bsums = (int*)alloc(4096 * 4);
    float* dinv = (float*)alloc((size_t)n_nodes * 4);
    int* nbr = (int*)alloc((size_t)twoE * 4);
    float* bufA = (float*)alloc((size_t)n_elems * 4);
    float* bufB = (float*)alloc((size_t)n_elems * 4);
    float* acc = (float*)d_out;

    const int T = 256;
    const int gN = (n_nodes + T - 1) / T;
    const int gE = (E + T - 1) / T;
    const int gElems = (n_elems + T - 1) / T;
    const int nblk = (n_nodes + 1023) / 1024;
    const int gLayer = (n_nodes + 7) / 8;  // 8 waves (nodes) per 256-thread block

    zero_int_kernel<<<gN, T, 0, stream>>>(deg, n_nodes);
    zero_int_kernel<<<gN, T, 0, stream>>>(fill, n_nodes);
    count_deg_kernel<<<gE, T, 0, stream>>>(user_ids, item_ids, deg, n_users, E);
    scan_block_kernel<<<nblk, 1024, 0, stream>>>(deg, offs, bsums, n_nodes);
    scan_sums_kernel<<<1, 32, 0, stream>>>(bsums, nblk);
    finalize_kernel<<<gN, T, 0, stream>>>(offs, bsums, deg, dinv, n_nodes, twoE);
    fill_csr_kernel<<<gE, T, 0, stream>>>(user_ids, item_ids, offs, fill, nbr, n_users, E);
    init_out_kernel<<<gElems, T, 0, stream>>>(user_emb, item_emb, acc, n_users, n_nodes);

    // layer 1: source = original embeddings (two separate tables)
    layer_kernel<<<gLayer, T, 0, stream>>>(user_emb, item_emb, bufA, acc,
                                           offs, nbr, dinv, n_users, n_nodes);
    // layer 2: source = bufA (contiguous node-major)
    layer_kernel<<<gLayer, T, 0, stream>>>(bufA, bufA + (size_t)n_users * EMBED_DIM, bufB, acc,
                                           offs, nbr, dinv, n_users, n_nodes);
    // layer 3: source = bufB, reuse bufA as sink
    layer_kernel<<<gLayer, T, 0, stream>>>(bufB, bufB + (size_t)n_users * EMBED_DIM, bufA, acc,
                                           offs, nbr, dinv, n_users, n_nodes);

    scale_kernel<<<gElems, T, 0, stream>>>(acc, 0.25f, n_elems);
    (void)n_in; (void)out_size; (void)ws_size; (void)n_items;
}